// MambaModel_83872121356679
// MI455X (gfx1250) — compile-verified
//
#include <hip/hip_runtime.h>
#include <hip/hip_bf16.h>
#include <math.h>

typedef float v2f __attribute__((ext_vector_type(2)));
typedef float v8f __attribute__((ext_vector_type(8)));
typedef unsigned int u32x4 __attribute__((ext_vector_type(4)));
typedef int i32x4 __attribute__((ext_vector_type(4)));
typedef int i32x8 __attribute__((ext_vector_type(8)));

#if defined(__has_builtin)
#if __has_builtin(__builtin_amdgcn_tensor_load_to_lds)
#define HAVE_TDM 1
#endif
#endif
#ifndef HAVE_TDM
#define HAVE_TDM 0
#endif

// ---- model dimensions (compile-time) ----
#define Bc   16
#define Lc   1080
#define Dc   128
#define NCc  64
#define DINc 256
#define NSTc 16
#define Kc   4
#define Rc   8
#define OUTc 3
#define NBc  3
#define BLc  (Bc * Lc)   // 17280, multiple of 16 and 64
#define KMAXc 256        // max K staged in LDS (32 x 256 f32 = 32 KB)

enum { EPI_NONE = 0, EPI_RELU = 1, EPI_BNRELU = 2 };

static __device__ __forceinline__ float silu_f(float x) {
  return x / (1.0f + expf(-x));
}

static __device__ __forceinline__ v8f wmma4(v2f a, v2f b, v8f c) {
  // V_WMMA_F32_16X16X4_F32 : D(16x16,f32) = A(16x4,f32) * B(4x16,f32) + C
  return __builtin_amdgcn_wmma_f32_16x16x4_f32(
      false, a, false, b, (short)0, c, false, false);
}

// ---------------- embedding + positional encoding ----------------
__global__ __launch_bounds__(256) void embed_k(const float* __restrict__ x,
                                               const float* __restrict__ eW,
                                               const float* __restrict__ eb,
                                               float* __restrict__ XE) {
  int idx = blockIdx.x * 256 + threadIdx.x;
  if (idx >= BLc * Dc) return;
  int d = idx & (Dc - 1);
  int r = idx >> 7;           // b*L + l
  int l = r % Lc;
  float v = x[r * 2 + 0] * eW[d] + x[r * 2 + 1] * eW[Dc + d] + eb[d];
  int j2 = d & ~1;
  float ang = (float)l * expf(-(float)j2 * (logf(10000.0f) / (float)Dc));
  v += (d & 1) ? cosf(ang) : sinf(ang);
  XE[idx] = v;
}

// ---------------- generic fp32-WMMA GEMM (ragged N; x-proj only) ----------------
// One wave per 16(M) x 32(N) tile: shared A fragment feeds 2 independent WMMA chains.
// Branchless N raggedness: clamp W row, scale fragment by 0/1 mask. M % 16 == 0.
template <int EPI>
__global__ __launch_bounds__(32) void gemm_wmma_k(
    const float* __restrict__ A, int lda, const float* __restrict__ W, int ldw,
    const float* __restrict__ bias, const float* __restrict__ bn,
    float* __restrict__ C, int ldc, int M, int N, int Kdim) {
  const int lane = threadIdx.x & 31;
  const int mr = lane & 15;
  const int kq = (lane >> 4) << 1;                 // K pair base: 0 or 2
  const int arow = blockIdx.x * 16 + mr;
  const int w0 = blockIdx.y * 32 + mr;
  const int w1 = w0 + 16;
  const float msk0 = (w0 < N) ? 1.0f : 0.0f;
  const float msk1 = (w1 < N) ? 1.0f : 0.0f;
  const int w0c = (w0 < N) ? w0 : (N - 1);
  const int w1c = (w1 < N) ? w1 : (N - 1);
  const float* ap = A + (size_t)arow * lda + kq;
  const float* wp0 = W + (size_t)w0c * ldw + kq;
  const float* wp1 = W + (size_t)w1c * ldw + kq;
  v8f acc0 = {}, acc1 = {};
#pragma unroll 2
  for (int k = 0; k < Kdim; k += 4) {
    if ((k & 31) == 0) __builtin_prefetch(ap + k + 32, 0, 0);
    v2f a = *(const v2f*)(ap + k);
    v2f b0 = *(const v2f*)(wp0 + k);
    v2f b1 = *(const v2f*)(wp1 + k);
    b0.x *= msk0; b0.y *= msk0;
    b1.x *= msk1; b1.y *= msk1;
    acc0 = wmma4(a, b0, acc0);
    acc1 = wmma4(a, b1, acc1);
  }
  const int mbase = blockIdx.x * 16 + ((lane >> 4) << 3);
#pragma unroll 2
  for (int half = 0; half < 2; ++half) {
    const v8f& acc = half ? acc1 : acc0;
    const int nc = blockIdx.y * 32 + half * 16 + (lane & 15);
    if (nc >= N) continue;
    float bv = bias ? bias[nc] : 0.0f;
    float bt = 0.0f, mn = 0.0f, sc = 1.0f;
    if (EPI == EPI_BNRELU) {
      bt = bn[N + nc]; mn = bn[2 * N + nc];
      sc = bn[nc] * rsqrtf(bn[3 * N + nc] + 1e-5f);
    }
#pragma unroll
    for (int r = 0; r < 8; ++r) {
      float v = acc[r] + bv;
      if (EPI == EPI_RELU)   v = fmaxf(v, 0.0f);
      if (EPI == EPI_BNRELU) v = fmaxf((v - mn) * sc + bt, 0.0f);
      C[(size_t)(mbase + r) * ldc + nc] = v;
    }
  }
}

// ---------------- TDM-staged GEMM: C = act(A * W^T + bias) ----------------
// 256 threads = 8 waves (4 M-tiles x 2 N-tiles) per 64(M) x 32(N) block tile.
// Weight tile (32 rows x Kdim) DMA'd to LDS once via the Tensor Data Mover,
// then all 8 waves read B fragments from LDS (ds_load_b64) while streaming A.
// Requires: Kdim <= KMAXc, Kdim % 8 == 0, N % 32 == 0, M % 64 == 0, ldw == Kdim.
template <int EPI>
__global__ __launch_bounds__(256) void gemm_wmma_tdm_k(
    const float* __restrict__ A, int lda, const float* __restrict__ W, int ldw,
    const float* __restrict__ bias, const float* __restrict__ bn,
    float* __restrict__ C, int ldc, int N, int Kdim) {
  __shared__ float sW[32 * KMAXc];
  const int tid = threadIdx.x;
  const int wave = tid >> 5, lane = tid & 31;
  const int mi = wave & 3, ni = wave >> 2;
  const int ntile0 = blockIdx.y * 32;
#if HAVE_TDM
  if (wave == 0) {
    // Tensor DMA descriptor (CDNA5 ISA):
    // 2D tile: tile_dim0 = Kdim elems/row, tile_dim1 = 32 rows, 4-byte elems.
    unsigned long long gaddr =
        (unsigned long long)(uintptr_t)(W + (size_t)ntile0 * ldw);
    // flat shared addr low 32 bits == LDS byte offset (aperture in [63:32])
    unsigned int ldsoff = (unsigned int)(uintptr_t)&sW[0];
    u32x4 g0;
    g0.x = 1u;                                        // count=1, user descriptor
    g0.y = ldsoff;                                    // lds_addr
    g0.z = (unsigned int)gaddr;                       // global_addr[31:0]
    g0.w = (unsigned int)((gaddr >> 32) & 0x1FFFFFFu) // global_addr[56:32]
           | 0x80000000u;                             // type=2 ("image")
    unsigned long long kd = (unsigned long long)(unsigned)Kdim;
    unsigned long long st = (unsigned long long)(unsigned)ldw;
    unsigned long long q0 = 0x20000ull                // data_size=2 (4 bytes)
                          | ((kd & 0xFFFFull) << 48); // tensor_dim0 low16
    unsigned long long q1 = ((kd >> 16) & 0xFFFFull)  // tensor_dim0 high16
                          | (32ull << 16)             // tensor_dim1 = 32 rows
                          | (kd << 48);               // tile_dim0 = Kdim
    unsigned long long q2 = 32ull                     // tile_dim1 = 32
                          | (st << 32);               // tensor_dim0_stride lo32
    unsigned long long q3 = (st >> 32) & 0xFFFFull;   // stride hi16; dim1_stride=0
    i32x8 g1;
    g1[0] = (int)(unsigned)q0; g1[1] = (int)(q0 >> 32);
    g1[2] = (int)(unsigned)q1; g1[3] = (int)(q1 >> 32);
    g1[4] = (int)(unsigned)q2; g1[5] = (int)(q2 >> 32);
    g1[6] = (int)(unsigned)q3; g1[7] = (int)(q3 >> 32);
    i32x4 z4 = {0, 0, 0, 0};
#if defined(__clang_major__) && (__clang_major__ >= 23)
    i32x8 z8 = {};
    __builtin_amdgcn_tensor_load_to_lds(g0, g1, z4, z4, z8, 0);
#else
    __builtin_amdgcn_tensor_load_to_lds(g0, g1, z4, z4, 0);
#endif
    __builtin_amdgcn_s_wait_tensorcnt((short)0);
  }
#else
  {
    const float* Wg = W + (size_t)ntile0 * ldw;
    for (int i = tid; i < 32 * Kdim; i += 256) sW[i] = Wg[i];
  }
#endif
  __syncthreads();
  const int mr = lane & 15;
  const int kq = (lane >> 4) << 1;
  const float* ap = A + (size_t)(blockIdx.x * 64 + mi * 16 + mr) * lda + kq;
  const float* bp = &sW[(ni * 16 + mr) * Kdim + kq];
  v8f accE = {}, accO = {};
  for (int k = 0; k < Kdim; k += 8) {
    v2f a0 = *(const v2f*)(ap + k);
    v2f a1 = *(const v2f*)(ap + k + 4);
    v2f b0 = *(const v2f*)(bp + k);       // ds_load_b64 from staged tile
    v2f b1 = *(const v2f*)(bp + k + 4);
    accE = wmma4(a0, b0, accE);
    accO = wmma4(a1, b1, accO);
  }
  const int mbase = blockIdx.x * 64 + mi * 16 + ((lane >> 4) << 3);
  const int nc = ntile0 + ni * 16 + (lane & 15);
  float bv = bias ? bias[nc] : 0.0f;
  float bt = 0.0f, mn = 0.0f, sc = 1.0f;
  if (EPI == EPI_BNRELU) {
    bt = bn[N + nc]; mn = bn[2 * N + nc];
    sc = bn[nc] * rsqrtf(bn[3 * N + nc] + 1e-5f);
  }
#pragma unroll
  for (int r = 0; r < 8; ++r) {
    float v = accE[r] + accO[r] + bv;
    if (EPI == EPI_RELU)   v = fmaxf(v, 0.0f);
    if (EPI == EPI_BNRELU) v = fmaxf((v - mn) * sc + bt, 0.0f);
    C[(size_t)(mbase + r) * ldc + nc] = v;
  }
}

// ---------------- conv1d (channels-last) as tap-accumulated WMMA GEMM ----------------
__global__ __launch_bounds__(32) void conv_bnrelu_wmma_k(
    const float* __restrict__ XE, const float* __restrict__ W,
    const float* __restrict__ cb, const float* __restrict__ bn,
    float* __restrict__ CC, int ldcc, int coff, int KS, int pad) {
  const int lane = threadIdx.x & 31;
  const int mr = lane & 15;
  const int kq = (lane >> 4) << 1;
  const int r = blockIdx.x * 16 + mr;   // b*L + l
  const int b = r / Lc;
  const int l = r % Lc;
  const int oc = blockIdx.y * 16 + mr;
  v8f accE = {}, accO = {};
  for (int t = 0; t < KS; ++t) {
    int ls = l + t - pad;
    float mf = (ls >= 0 && ls < Lc) ? 1.0f : 0.0f;
    int lsc = ls < 0 ? 0 : (ls >= Lc ? Lc - 1 : ls);
    const float* ap = XE + (size_t)(b * Lc + lsc) * Dc + kq;
    const float* wp = W + ((size_t)oc * Dc + kq) * KS + t;  // channel stride KS
#pragma unroll 2
    for (int k = 0; k < Dc; k += 8) {
      v2f a0 = *(const v2f*)(ap + k);
      v2f a1 = *(const v2f*)(ap + k + 4);
      a0.x *= mf; a0.y *= mf; a1.x *= mf; a1.y *= mf;
      v2f w0, w1;
      w0.x = wp[(size_t)k * KS];
      w0.y = wp[(size_t)(k + 1) * KS];
      w1.x = wp[(size_t)(k + 4) * KS];
      w1.y = wp[(size_t)(k + 5) * KS];
      accE = wmma4(a0, w0, accE);
      accO = wmma4(a1, w1, accO);
    }
  }
  const int nc = blockIdx.y * 16 + (lane & 15);
  const int mbase = blockIdx.x * 16 + ((lane >> 4) << 3);
  float bv = cb[nc];
  float bt = bn[NCc + nc], mn = bn[2 * NCc + nc];
  float sc = bn[nc] * rsqrtf(bn[3 * NCc + nc] + 1e-5f);
#pragma unroll
  for (int rr = 0; rr < 8; ++rr) {
    float v = accE[rr] + accO[rr] + bv;
    v = fmaxf((v - mn) * sc + bt, 0.0f);
    CC[(size_t)(mbase + rr) * ldcc + coff + nc] = v;
  }
}

// ---------------- layernorm: one wave per 128-wide row ----------------
__global__ __launch_bounds__(128) void ln_k(const float* __restrict__ X,
                                            const float* __restrict__ g,
                                            const float* __restrict__ bta,
                                            float* __restrict__ Yo) {
  int wid = threadIdx.x >> 5, lane = threadIdx.x & 31;
  int row = blockIdx.x * 4 + wid;
  int d = lane * 4;
  float4 v = *(const float4*)(X + (size_t)row * Dc + d);
  float s = v.x + v.y + v.z + v.w;
#pragma unroll
  for (int m = 16; m; m >>= 1) s += __shfl_xor(s, m, 32);
  float mu = s * (1.0f / (float)Dc);
  float dx = v.x - mu, dy = v.y - mu, dz = v.z - mu, dw = v.w - mu;
  float q = dx * dx + dy * dy + dz * dz + dw * dw;
#pragma unroll
  for (int m = 16; m; m >>= 1) q += __shfl_xor(q, m, 32);
  float rs = rsqrtf(q * (1.0f / (float)Dc) + 1e-5f);
  float4 o;
  o.x = dx * rs * g[d + 0] + bta[d + 0];
  o.y = dy * rs * g[d + 1] + bta[d + 1];
  o.z = dz * rs * g[d + 2] + bta[d + 2];
  o.w = dw * rs * g[d + 3] + bta[d + 3];
  *(float4*)(Yo + (size_t)row * Dc + d) = o;
}

// ---------------- depthwise causal conv (K=4) + SiLU ----------------
__global__ __launch_bounds__(256) void dwconv_silu_k(const float* __restrict__ XZ,
                                                     const float* __restrict__ cw,
                                                     const float* __restrict__ cbias,
                                                     float* __restrict__ U) {
  int idx = blockIdx.x * 256 + threadIdx.x;     // grid = BLc blocks
  int d = idx & (DINc - 1);
  int r = idx >> 8;
  int l = r % Lc;
  float acc = cbias[d];
#pragma unroll
  for (int t = 0; t < Kc; ++t) {
    int ls = l - (Kc - 1) + t;
    if (ls >= 0)
      acc = fmaf(cw[d * Kc + t], XZ[(size_t)(r - (Kc - 1) + t) * (2 * DINc) + d], acc);
  }
  U[idx] = silu_f(acc);
}

// ---------------- dt projection (K=8) + softplus ----------------
__global__ __launch_bounds__(256) void dtproj_k(const float* __restrict__ DBL,
                                                const float* __restrict__ dtW,
                                                const float* __restrict__ dtb,
                                                float* __restrict__ DTo) {
  int idx = blockIdx.x * 256 + threadIdx.x;
  int d = idx & (DINc - 1);
  int r = idx >> 8;
  const float* row = DBL + (size_t)r * 40;
  float s = dtb[d];
#pragma unroll
  for (int j = 0; j < Rc; ++j) s = fmaf(row[j], dtW[d * Rc + j], s);
  DTo[idx] = (s > 20.0f) ? s : log1pf(expf(s));
}

// ---------------- selective scan: thread = one (b,d) channel, h[16] in VGPRs ----------------
__global__ __launch_bounds__(256) void scan_k(const float* __restrict__ DT,
                                              const float* __restrict__ U,
                                              const float* __restrict__ DBL,
                                              const float* __restrict__ Alog,
                                              const float* __restrict__ Dp,
                                              float* __restrict__ Y) {
  __shared__ float sB[NSTc], sC[NSTc];
  const int b = blockIdx.x;
  const int d = threadIdx.x;
  float A[NSTc], h[NSTc];
#pragma unroll
  for (int n = 0; n < NSTc; ++n) {
    A[n] = -expf(Alog[d * NSTc + n]);
    h[n] = 0.0f;
  }
  const float dpv = Dp[d];
  for (int l = 0; l < Lc; ++l) {
    const int r = b * Lc + l;
    if (threadIdx.x < NSTc)
      sB[threadIdx.x] = DBL[(size_t)r * 40 + Rc + threadIdx.x];
    else if (threadIdx.x < 2 * NSTc)
      sC[threadIdx.x - NSTc] = DBL[(size_t)r * 40 + Rc + NSTc + (threadIdx.x - NSTc)];
    __syncthreads();
    const float dt = DT[(size_t)r * DINc + d];
    const float u = U[(size_t)r * DINc + d];
    const float du = dt * u;
    float y = 0.0f;
#pragma unroll
    for (int n = 0; n < NSTc; ++n) {
      float dA = __expf(dt * A[n]);
      h[n] = fmaf(dA, h[n], du * sB[n]);
      y = fmaf(h[n], sC[n], y);
    }
    Y[(size_t)r * DINc + d] = fmaf(u, dpv, y);
    __syncthreads();
  }
}

// ---------------- gating: G = y * silu(z) ----------------
__global__ __launch_bounds__(256) void gate_k(const float* __restrict__ Y,
                                              const float* __restrict__ XZ,
                                              float* __restrict__ G) {
  int idx = blockIdx.x * 256 + threadIdx.x;
  int d = idx & (DINc - 1);
  int r = idx >> 8;
  G[idx] = Y[idx] * silu_f(XZ[(size_t)r * (2 * DINc) + DINc + d]);
}

// ---------------- concat(cnn_feat, m) ----------------
__global__ __launch_bounds__(256) void concat_k(const float* __restrict__ CNNF,
                                                const float* __restrict__ M,
                                                float* __restrict__ FIN) {
  int idx = blockIdx.x * 256 + threadIdx.x;
  int d = idx & 255;
  int r = idx >> 8;
  FIN[idx] = (d < Dc) ? CNNF[(size_t)r * Dc + d] : M[(size_t)r * Dc + (d - Dc)];
}

// ---------------- mean pool over L ----------------
__global__ __launch_bounds__(256) void pool_k(const float* __restrict__ FU,
                                              float* __restrict__ P) {
  int idx = blockIdx.x * 256 + threadIdx.x;
  if (idx >= Bc * Dc) return;
  int b = idx >> 7, d = idx & 127;
  float s = 0.0f;
  for (int l = 0; l < Lc; ++l) s += FU[(size_t)(b * Lc + l) * Dc + d];
  P[idx] = s * (1.0f / (float)Lc);
}

// ---------------- final classifier head ----------------
__global__ __launch_bounds__(64) void fc_k(const float* __restrict__ P,
                                           const float* __restrict__ W,
                                           const float* __restrict__ bb,
                                           float* __restrict__ out) {
  int idx = threadIdx.x;
  if (idx >= Bc * OUTc) return;
  int b = idx / OUTc, o = idx % OUTc;
  float s = bb[o];
  for (int d = 0; d < Dc; ++d) s = fmaf(P[b * Dc + d], W[o * Dc + d], s);
  out[idx] = s;
}

extern "C" void kernel_launch(void* const* d_in, const int* in_sizes, int n_in,
                              void* d_out, int out_size, void* d_ws, size_t ws_size,
                              hipStream_t stream) {
  const float* x    = (const float*)d_in[0];
  const float* embW = (const float*)d_in[1];
  const float* embb = (const float*)d_in[2];
  const float* c3W  = (const float*)d_in[3];
  const float* c3b  = (const float*)d_in[4];
  const float* bn3  = (const float*)d_in[5];
  const float* c5W  = (const float*)d_in[6];
  const float* c5b  = (const float*)d_in[7];
  const float* bn5  = (const float*)d_in[8];
  const float* c7W  = (const float*)d_in[9];
  const float* c7b  = (const float*)d_in[10];
  const float* bn7  = (const float*)d_in[11];
  const float* cfW  = (const float*)d_in[12];
  const float* cfb  = (const float*)d_in[13];
  const float* bnf  = (const float*)d_in[14];
  const float* lng  = (const float*)d_in[15];
  const float* lnb  = (const float*)d_in[16];
  const float* inW  = (const float*)d_in[17];
  const float* mcW  = (const float*)d_in[18];
  const float* mcb  = (const float*)d_in[19];
  const float* xpW  = (const float*)d_in[20];
  const float* dtW  = (const float*)d_in[21];
  const float* dtb  = (const float*)d_in[22];
  const float* Alog = (const float*)d_in[23];
  const float* Dpp  = (const float*)d_in[24];
  const float* outW = (const float*)d_in[25];
  const float* flW  = (const float*)d_in[26];
  const float* flb  = (const float*)d_in[27];
  const float* fcW  = (const float*)d_in[28];
  const float* fcb  = (const float*)d_in[29];
  float* out = (float*)d_out;
  (void)in_sizes; (void)n_in; (void)out_size; (void)ws_size;

  // workspace layout (fp32), buffers aliased where lifetimes allow
  float* ws = (float*)d_ws;
  size_t o = 0;
  float* XE   = ws + o; o += (size_t)BLc * Dc;
  float* M0   = ws + o; o += (size_t)BLc * Dc;
  float* M1   = ws + o; o += (size_t)BLc * Dc;
  float* CNNF = ws + o; o += (size_t)BLc * Dc;
  float* LN   = ws + o; o += (size_t)BLc * Dc;
  float* XZ   = ws + o; o += (size_t)BLc * 2 * DINc;  // also CC, FIN
  float* Ub   = ws + o; o += (size_t)BLc * DINc;      // also FUSED
  float* DTb  = ws + o; o += (size_t)BLc * DINc;      // also G
  float* DBL  = ws + o; o += (size_t)BLc * 40;
  float* Yb   = ws + o; o += (size_t)BLc * DINc;
  float* PO   = ws + o; o += (size_t)Bc * Dc;
  float* CC = XZ;
  float* FIN = XZ;
  float* Gb = DTb;
  float* FU = Ub;

  dim3 b32(32), b128(128), b256(256);

  // 1) embedding + PE
  embed_k<<<dim3((BLc * Dc + 255) / 256), b256, 0, stream>>>(x, embW, embb, XE);

  // 2) CNN branch: conv{3,5,7} (WMMA, fused BN+ReLU, channels-last),
  //    then 1x1 fuse conv as TDM-staged GEMM (K=192) + BN + ReLU
  conv_bnrelu_wmma_k<<<dim3(BLc / 16, NCc / 16), b32, 0, stream>>>(XE, c3W, c3b, bn3, CC, 3 * NCc, 0, 3, 1);
  conv_bnrelu_wmma_k<<<dim3(BLc / 16, NCc / 16), b32, 0, stream>>>(XE, c5W, c5b, bn5, CC, 3 * NCc, NCc, 5, 2);
  conv_bnrelu_wmma_k<<<dim3(BLc / 16, NCc / 16), b32, 0, stream>>>(XE, c7W, c7b, bn7, CC, 3 * NCc, 2 * NCc, 7, 3);
  gemm_wmma_tdm_k<EPI_BNRELU><<<dim3(BLc / 64, Dc / 32), b256, 0, stream>>>(
      CC, 3 * NCc, cfW, 3 * NCc, cfb, bnf, CNNF, Dc, Dc, 3 * NCc);

  // 3) Mamba blocks
  const float* Min = XE;
  float* Mout = M0;
  for (int i = 0; i < NBc; ++i) {
    ln_k<<<dim3(BLc / 4), b128, 0, stream>>>(Min, lng + i * Dc, lnb + i * Dc, LN);
    // in-proj: (17280 x 512) = LN(17280x128) @ inW^T  (TDM-staged weights)
    gemm_wmma_tdm_k<EPI_NONE><<<dim3(BLc / 64, (2 * DINc) / 32), b256, 0, stream>>>(
        LN, Dc, inW + (size_t)i * 2 * DINc * Dc, Dc, nullptr, nullptr,
        XZ, 2 * DINc, 2 * DINc, Dc);
    dwconv_silu_k<<<dim3(BLc), b256, 0, stream>>>(XZ, mcW + i * DINc * Kc, mcb + i * DINc, Ub);
    // x-proj: ragged N=40 -> per-wave masked GEMM
    gemm_wmma_k<EPI_NONE><<<dim3(BLc / 16, 2), b32, 0, stream>>>(
        Ub, DINc, xpW + (size_t)i * (Rc + 2 * NSTc) * DINc, DINc, nullptr, nullptr,
        DBL, 40, BLc, 40, DINc);
    dtproj_k<<<dim3(BLc), b256, 0, stream>>>(DBL, dtW + i * DINc * Rc, dtb + i * DINc, DTb);
    scan_k<<<dim3(Bc), b256, 0, stream>>>(DTb, Ub, DBL,
                                          Alog + (size_t)i * DINc * NSTc, Dpp + i * DINc, Yb);
    gate_k<<<dim3(BLc), b256, 0, stream>>>(Yb, XZ, Gb);
    // out-proj: (17280 x 128), K=256 (TDM-staged weights)
    gemm_wmma_tdm_k<EPI_NONE><<<dim3(BLc / 64, Dc / 32), b256, 0, stream>>>(
        Gb, DINc, outW + (size_t)i * Dc * DINc, DINc, nullptr, nullptr,
        Mout, Dc, Dc, DINc);
    Min = Mout;
    Mout = (i == 0) ? M1 : M0;
  }

  // 4) fusion: concat -> linear+ReLU (TDM-staged, K=256) -> mean pool -> head
  concat_k<<<dim3(BLc), b256, 0, stream>>>(CNNF, Min, FIN);
  gemm_wmma_tdm_k<EPI_RELU><<<dim3(BLc / 64, Dc / 32), b256, 0, stream>>>(
      FIN, 2 * Dc, flW, 2 * Dc, flb, nullptr, FU, Dc, Dc, 2 * Dc);
  pool_k<<<dim3((Bc * Dc + 255) / 256), b256, 0, stream>>>(FU, PO);
  fc_k<<<dim3(1), dim3(64), 0, stream>>>(PO, fcW, fcb, out);
}